// RWW_Layer_75058848464968
// MI455X (gfx1250) — compile-verified
//
#include <hip/hip_runtime.h>
#include <stdint.h>

// ---------------- configuration ----------------
#define NWG      25          // workgroups; NWG*ROWS == NR
#define BLOCK    256         // 8 wave32s
#define ROWS     32          // matrix rows owned per workgroup
#define NR       800         // regions
#define MSTRIDE  804         // padded LDS row stride (floats): even, 16B rows, conflict-free

typedef float v2f __attribute__((ext_vector_type(2)));
typedef float v8f __attribute__((ext_vector_type(8)));

// LDS byte offset of a generic pointer to __shared__ (low 32 bits of generic addr)
__device__ __forceinline__ uint32_t lds_off(const void* p) {
    return (uint32_t)(uintptr_t)p;
}

// CDNA5 async global->LDS copy (tracked by ASYNCcnt)
__device__ __forceinline__ void async_b128_to_lds(void* lds_dst, const void* gsrc) {
    asm volatile("global_load_async_to_lds_b128 %0, %1, off"
                 :: "v"(lds_off(lds_dst)), "v"(gsrc)
                 : "memory");
}

__device__ __forceinline__ void wait_async0() {
#if __has_builtin(__builtin_amdgcn_s_wait_asynccnt)
    __builtin_amdgcn_s_wait_asynccnt(0);
#else
    asm volatile("s_wait_asynccnt 0x0" ::: "memory");
#endif
}

// exact fp32 replica of the reference _H
__device__ __forceinline__ float Hfun(float I, float a, float b, float d, float big) {
    float x = a * I - b;
    float numer = fabsf(x) + 1e-9f;
    float z = -d * x;
    float denom;
    if (z > 50.0f) denom = fabsf(1.0f - big * z) + 1e-9f * d;
    else           denom = fabsf(1.0f - expf(fminf(z, 51.0f))) + 1e-9f * d;
    return numer / denom;
}

__global__ void rww_init(int* bar) {
    if (threadIdx.x == 0) *bar = 0;
}

__global__ void __launch_bounds__(BLOCK, 1)
rww_persistent(const float* __restrict__ Con, const float* __restrict__ init_state,
               const float* __restrict__ v_of_T, const int* __restrict__ pnum,
               float* __restrict__ out, float* __restrict__ gbuf, int* __restrict__ bar)
{
    extern __shared__ float smem[];
    float* mat   = smem;                      // ROWS * MSTRIDE
    float* sE    = smem + ROWS * MSTRIDE;     // NR   (16B aligned: 25728*4 % 16 == 0)
    float* sI    = sE + NR;                   // ROWS
    float* cpart = sI + ROWS;                 // 8 waves * 16 rows

    const int tid  = threadIdx.x;
    const int wg   = blockIdx.x;
    const int lane = tid & 31;
    const int wave = tid >> 5;
    const int row0 = wg * ROWS;
    const int nsteps = *pnum;

    // ---- stage this WG's Con_Mtx row slice into LDS (async DMA path) ----
    for (int idx = tid; idx < ROWS * (NR / 4); idx += BLOCK) {
        int rl = idx / (NR / 4);
        int jc = (idx - rl * (NR / 4)) * 4;
        async_b128_to_lds(mat + rl * MSTRIDE + jc,
                          Con + (size_t)(row0 + rl) * NR + jc);
    }
    for (int i = tid; i < NR; i += BLOCK) sE[i] = init_state[2 * i];
    if (tid < ROWS) sI[tid] = init_state[2 * (row0 + tid) + 1];
    wait_async0();
    __syncthreads();

    // wave -> (16-row tile, 200-wide K slice)
    const int tileR = (wave & 1) * 16;
    const int kpart = wave >> 1;
    const int rloc  = tileR + (lane & 15);
    const int ksel  = (lane >> 4) * 2;        // lanes 0-15: K+0/1, lanes 16-31: K+2/3
    const float* matrow = mat + rloc * MSTRIDE;

    for (int t = 0; t < nsteps; ++t) {
        // ---- coupling = Con_slice @ S_E via fp32 WMMA (broadcast-B trick) ----
#if __has_builtin(__builtin_amdgcn_wmma_f32_16x16x4_f32)
        v8f acc0 = {}, acc1 = {};
        const int kbase = kpart * 200;
        #pragma unroll 5
        for (int i = 0; i < 50; i += 2) {
            int k0 = kbase + i * 4 + ksel;
            v2f a0 = *(const v2f*)(matrow + k0);
            v2f b0 = *(const v2f*)(sE + k0);
            acc0 = __builtin_amdgcn_wmma_f32_16x16x4_f32(false, a0, false, b0,
                                                         (short)0, acc0, false, false);
            int k1 = k0 + 4;
            v2f a1 = *(const v2f*)(matrow + k1);
            v2f b1 = *(const v2f*)(sE + k1);
            acc1 = __builtin_amdgcn_wmma_f32_16x16x4_f32(false, a1, false, b1,
                                                         (short)0, acc1, false, false);
        }
        v8f acc = acc0 + acc1;
        // D layout: VGPR r, lanes 0-15 -> M=r, lanes 16-31 -> M=r+8 (all N equal here)
        if ((lane & 15) == 0) {
            float* dst = cpart + wave * 16 + (lane >> 4) * 8;
            #pragma unroll
            for (int r = 0; r < 8; ++r) dst[r] = acc[r];
        }
#else
        // VALU fallback: same cpart contract
        float p = 0.0f;
        const int kbase = kpart * 200;
        #pragma unroll 10
        for (int i = 0; i < 50; ++i) {
            int k = kbase + i * 4 + ksel;
            v2f a = *(const v2f*)(matrow + k);
            v2f b = *(const v2f*)(sE + k);
            p += a.x * b.x + a.y * b.y;
        }
        p += __shfl_xor(p, 16);
        if (lane < 16) cpart[wave * 16 + lane] = p;
#endif
        __syncthreads();

        // ---- pointwise state update on owner threads ----
        if (tid < ROWS) {
            int region = row0 + tid;
            int tile = tid >> 4, rl = tid & 15;
            float y = cpart[(0 * 2 + tile) * 16 + rl] + cpart[(1 * 2 + tile) * 16 + rl]
                    + cpart[(2 * 2 + tile) * 16 + rl] + cpart[(3 * 2 + tile) * 16 + rl];
            float se = sE[region];
            float si = sI[tid];
            float vn = v_of_T[(size_t)t * NR + region];

            float IE = 0.382f + 0.21f * se + 3.0f * y - si;        // W_E*I0 + W+*Jn*SE + G*Jn*coup - J*SI
            float II = 0.2674f + 0.15f * se - si;                  // W_I*I0 + Jn*SE - SI (lambda=0)
            float rE = Hfun(IE, 310.0f, 125.0f, 0.16f,  1.0e9f);
            float rI = Hfun(II, 615.0f, 177.0f, 0.087f, 1.0e5f);
            float dse = -se * 0.01f + (1.0f - se) * 0.000641f * rE + 0.01f * vn;
            float dsi = -si * 0.1f  + 0.001f * rI + 0.01f * vn;
            se += 0.05f * dse;
            si += 0.05f * dsi;

            sI[tid] = si;
            gbuf[((t + 1) & 1) * NR + region] = se;                // publish new S_E
            v2f hv = {se, si};
            *(v2f*)(out + 2 * NR + 2 * ((size_t)t * NR + region)) = hv;   // state_hist
            if (t + 1 < nsteps)
                __builtin_prefetch(v_of_T + (size_t)(t + 1) * NR + region, 0, 1);
        }

        // ---- device-wide step barrier (one per step; S_E ping-pong removes WAR) ----
        __threadfence();
        __syncthreads();
#if __has_builtin(__builtin_amdgcn_s_cluster_barrier)
        __builtin_amdgcn_s_cluster_barrier();   // NOP when not dispatched as a cluster
#endif
        if (tid == 0) {
            __hip_atomic_fetch_add(bar, 1, __ATOMIC_RELEASE, __HIP_MEMORY_SCOPE_AGENT);
            int target = (t + 1) * NWG;
            while (__hip_atomic_load(bar, __ATOMIC_ACQUIRE, __HIP_MEMORY_SCOPE_AGENT) < target)
                __builtin_amdgcn_s_sleep(1);
        }
        __syncthreads();

        // ---- async-refresh full S_E broadcast into LDS ----
        {
            const float* src = gbuf + ((t + 1) & 1) * NR;
            if (tid < NR / 4)
                async_b128_to_lds(sE + tid * 4, src + tid * 4);
            wait_async0();
        }
        __syncthreads();
    }

    // ---- final state_vals (S_E, S_I) per region ----
    if (tid < ROWS) {
        int region = row0 + tid;
        out[2 * region]     = sE[region];
        out[2 * region + 1] = sI[tid];
    }
}

extern "C" void kernel_launch(void* const* d_in, const int* in_sizes, int n_in,
                              void* d_out, int out_size, void* d_ws, size_t ws_size,
                              hipStream_t stream) {
    const float* Con  = (const float*)d_in[0];   // (800,800)
    const float* init = (const float*)d_in[1];   // (800,2)
    const float* vT   = (const float*)d_in[2];   // (40000,800)
    const int*   pnum = (const int*)d_in[3];     // scalar num_steps

    float* out  = (float*)d_out;                 // [state_vals 1600 | state_hist 64M]
    float* gbuf = (float*)d_ws;                  // 2*800 float ping-pong S_E
    int*   bar  = (int*)((char*)d_ws + 2 * NR * sizeof(float));

    rww_init<<<1, 1, 0, stream>>>(bar);

    size_t shmem = (size_t)(ROWS * MSTRIDE + NR + ROWS + 8 * 16) * sizeof(float);
    rww_persistent<<<NWG, BLOCK, shmem, stream>>>(Con, init, vT, pnum, out, gbuf, bar);
    (void)in_sizes; (void)n_in; (void)out_size; (void)ws_size;
}